// MambaBlock_18150531793031
// MI455X (gfx1250) — compile-verified
//
#include <hip/hip_runtime.h>
#include <hip/hip_bf16.h>

// ---------------------------------------------------------------------------
// Mamba block for MI455X (gfx1250, wave32).
//  - GEMMs: fp32 WMMA (v_wmma_f32_16x16x4_f32), 2x2 register blocking per
//    wave, double-buffered LDS, global->reg->LDS software pipeline, B tile
//    swizzled in LDS so each WMMA fragment is one aligned ds_load_b64.
//  - Selective scan: 3-phase chunked parallel scan (64 chunks of 64 steps).
// ---------------------------------------------------------------------------

#define BSZ      4
#define LSEQ     4096
#define DMODEL   1024
#define DINNER   2048
#define DSTATE   16
#define DTRANK   64
#define DCONV    4
#define MROWS    (BSZ * LSEQ)          // 16384
#define NCHUNK   64
#define CLEN     64                    // NCHUNK * CLEN == LSEQ

typedef float v2f __attribute__((ext_vector_type(2)));
typedef float v8f __attribute__((ext_vector_type(8)));

__device__ __forceinline__ float softplus_f(float x) {
    return (x > 20.f) ? x : log1pf(__expf(x));
}
__device__ __forceinline__ float silu_f(float x) {
    return x / (1.f + __expf(-x));
}

// ---------------------------------------------------------------------------
// fp32 WMMA GEMM:  C[M x N] = A[M x K] * W[N x K]^T
// 256 threads = 8 waves in a 4(M) x 2(N) grid; each wave owns TM x TN tiles
// of 16x16  =>  block tile (64*TM) x (32*TN).  K chunked by 16, LDS double-
// buffered, next chunk staged in registers while current chunk computes.
// B tile is swizzled: K index k -> (kg=k>>2, hf=k&1, id2=(k>>1)&1) so that a
// lane's WMMA B fragment {K=kk+half, K=kk+half+2} is a contiguous float2.
// MODE 0: out[row*ldc + col] = v
// MODE 1: xz split: col <  DINNER -> out  (xh_pre, (B, DINNER, L) for conv)
//                   col >= DINNER -> out2 (res,   (B*L, DINNER))
// MODE 2: out[row*DINNER+col] = softplus(v + 2*bias[col])
// ---------------------------------------------------------------------------
template <int MODE, int TM, int TN>
__global__ void wmma_gemm_kernel(const float* __restrict__ A,
                                 const float* __restrict__ W,
                                 float* __restrict__ out,
                                 float* __restrict__ out2,
                                 const float* __restrict__ bias,
                                 int M, int N, int K,
                                 int lda, int ldw, int ldc) {
    constexpr int BM = 64 * TM;
    constexpr int BN = 32 * TN;
    __shared__ float As[2][BM][18];           // [buf][m][k]; pad 18 keeps 8B align
    __shared__ float Bsw[2][4][2][BN][2];     // [buf][kg][hf][n][id2]

    const int tid  = threadIdx.x;
    const int lane = tid & 31;
    const int wave = tid >> 5;            // 0..7
    const int wm   = wave & 3;            // 4 waves along M
    const int wn   = wave >> 2;           // 2 waves along N
    const int half = lane >> 4;           // 0 | 1
    const int q    = lane & 15;

    const int rowBase = blockIdx.y * BM;
    const int colBase = blockIdx.x * BN;

    v8f acc[TM][TN];
#pragma unroll
    for (int tm = 0; tm < TM; ++tm)
#pragma unroll
        for (int tn = 0; tn < TN; ++tn)
#pragma unroll
            for (int i = 0; i < 8; ++i) acc[tm][tn][i] = 0.f;

    float4 aReg[TM];
    float2 bReg[TN];

    auto loadG = [&](int k0) {
#pragma unroll
        for (int i = 0; i < TM; ++i) {
            const int v = tid + i * 256;          // BM*4 float4s total
            const int row = v >> 2, seg = v & 3;
            aReg[i] = *reinterpret_cast<const float4*>(
                A + (size_t)(rowBase + row) * lda + k0 + seg * 4);
        }
#pragma unroll
        for (int i = 0; i < TN; ++i) {
            const int v = tid + i * 256;          // BN*8 float2s total
            const int n = v >> 3, kp = (v & 7) * 2;
            bReg[i] = *reinterpret_cast<const float2*>(
                W + (size_t)(colBase + n) * ldw + k0 + kp);
        }
    };
    auto storeL = [&](int buf) {
#pragma unroll
        for (int i = 0; i < TM; ++i) {
            const int v = tid + i * 256;
            const int row = v >> 2, seg = v & 3;
            As[buf][row][seg * 4 + 0] = aReg[i].x;
            As[buf][row][seg * 4 + 1] = aReg[i].y;
            As[buf][row][seg * 4 + 2] = aReg[i].z;
            As[buf][row][seg * 4 + 3] = aReg[i].w;
        }
#pragma unroll
        for (int i = 0; i < TN; ++i) {
            const int v = tid + i * 256;
            const int n = v >> 3, kp = (v & 7) * 2;
            // swizzled B store: k -> [k>>2][k&1][n][(k>>1)&1]
            const int k0e = kp, k1e = kp + 1;
            Bsw[buf][k0e >> 2][k0e & 1][n][(k0e >> 1) & 1] = bReg[i].x;
            Bsw[buf][k1e >> 2][k1e & 1][n][(k1e >> 1) & 1] = bReg[i].y;
        }
    };
    auto compute = [&](int buf) {
#pragma unroll
        for (int kk = 0; kk < 16; kk += 4) {
            v2f af[TM], bf[TN];
#pragma unroll
            for (int tm = 0; tm < TM; ++tm) {
                const int row = (wm * TM + tm) * 16 + q;
                // contiguous, 8B-aligned pair -> single ds_load_b64
                const float2 av = *reinterpret_cast<const float2*>(
                    &As[buf][row][kk + 2 * half]);
                af[tm][0] = av.x;
                af[tm][1] = av.y;
            }
#pragma unroll
            for (int tn = 0; tn < TN; ++tn) {
                const int nc = (wn * TN + tn) * 16 + q;
                const float2 bv = *reinterpret_cast<const float2*>(
                    &Bsw[buf][kk >> 2][half][nc][0]);
                bf[tn][0] = bv.x;   // K = kk + half
                bf[tn][1] = bv.y;   // K = kk + half + 2
            }
#pragma unroll
            for (int tm = 0; tm < TM; ++tm)
#pragma unroll
                for (int tn = 0; tn < TN; ++tn)
                    acc[tm][tn] = __builtin_amdgcn_wmma_f32_16x16x4_f32(
                        false, af[tm], false, bf[tn], (short)0,
                        acc[tm][tn], false, false);
        }
    };

    // software pipeline: reg-stage chunk k+1 while computing chunk k
    loadG(0);
    storeL(0);
    __syncthreads();
    int buf = 0;
    for (int k0 = 16; k0 < K; k0 += 16) {
        loadG(k0);           // in flight during compute below
        compute(buf);
        storeL(buf ^ 1);     // loadcnt wait lands here, after the WMMAs
        __syncthreads();
        buf ^= 1;
    }
    compute(buf);

    // Epilogue.  C/D layout: VGPR i -> M = i + 8*half, N = lane&15.
#pragma unroll
    for (int tm = 0; tm < TM; ++tm) {
#pragma unroll
        for (int tn = 0; tn < TN; ++tn) {
#pragma unroll
            for (int i = 0; i < 8; ++i) {
                const int m   = (wm * TM + tm) * 16 + 8 * half + i;
                const int n   = (wn * TN + tn) * 16 + q;
                const int row = rowBase + m;
                const int col = colBase + n;
                const float v = acc[tm][tn][i];
                if (MODE == 0) {
                    out[(size_t)row * ldc + col] = v;
                } else if (MODE == 1) {
                    const int b = row >> 12;           // row / LSEQ
                    const int l = row & (LSEQ - 1);
                    if (col < DINNER) {
                        out[((size_t)b * DINNER + col) * LSEQ + l] = v;
                    } else {
                        out2[(size_t)row * DINNER + (col - DINNER)] = v;
                    }
                } else { // MODE 2
                    out[(size_t)row * DINNER + col] =
                        softplus_f(v + 2.f * bias[col]);
                }
            }
        }
    }
}

// ---------------------------------------------------------------------------
// Causal depthwise conv (width 4) + bias + SiLU.
// In : xh_pre  (B, DINNER, L)   Out: xh_act (B, L, DINNER)
// ---------------------------------------------------------------------------
__global__ void conv_silu_kernel(const float* __restrict__ xh_pre,
                                 const float* __restrict__ conv_w,
                                 const float* __restrict__ conv_b,
                                 float* __restrict__ xh_act) {
    const size_t g = (size_t)blockIdx.x * blockDim.x + threadIdx.x;
    const int l = (int)(g & (LSEQ - 1));
    const size_t rest = g >> 12;
    const int d = (int)(rest & (DINNER - 1));
    const int b = (int)(rest >> 11);

    const float* src = xh_pre + ((size_t)b * DINNER + d) * LSEQ;
    const float w0 = conv_w[d * DCONV + 0];
    const float w1 = conv_w[d * DCONV + 1];
    const float w2 = conv_w[d * DCONV + 2];
    const float w3 = conv_w[d * DCONV + 3];

    float acc = conv_b[d];
    if (l >= 3) acc = fmaf(src[l - 3], w0, acc);
    if (l >= 2) acc = fmaf(src[l - 2], w1, acc);
    if (l >= 1) acc = fmaf(src[l - 1], w2, acc);
    acc = fmaf(src[l], w3, acc);

    xh_act[((size_t)b * LSEQ + l) * DINNER + d] = silu_f(acc);
}

// ---------------------------------------------------------------------------
// Chunked parallel selective scan.  Operator s <- dA*s + dBu is associative:
//   phase A: per (b,d,chunk) compute (prod dA, local end-state)   [parallel]
//   phase B: per (b,d) 64-step sequential combine -> chunk init states
//   phase C: per (b,d,chunk) replay from init state, emit gated y [parallel]
// One lane per chain; 16 states in registers; lanes of a wave share (b,l)
// so the 32 B/C values are fetched one-per-lane and broadcast via __shfl.
// ---------------------------------------------------------------------------
__global__ void scan_phase_a(const float* __restrict__ u,
                             const float* __restrict__ delta,
                             const float* __restrict__ x_dbl,
                             const float* __restrict__ A_log,
                             float* __restrict__ chunkA,
                             float* __restrict__ chunkS) {
    const int g = blockIdx.x * blockDim.x + threadIdx.x;  // 0..524287
    const int d = g & (DINNER - 1);
    const int rest = g >> 11;
    const int c = rest & (NCHUNK - 1);
    const int b = rest >> 6;
    const int lane = threadIdx.x & 31;

    float Arow[DSTATE], aP[DSTATE], s[DSTATE];
#pragma unroll
    for (int n = 0; n < DSTATE; ++n) {
        Arow[n] = -__expf(A_log[d * DSTATE + n]);
        aP[n] = 1.f;
        s[n]  = 0.f;
    }

    const size_t rowBase = (size_t)b * LSEQ + (size_t)c * CLEN;
    for (int l0 = 0; l0 < CLEN; ++l0) {
        const size_t r = rowBase + l0;
        const float dlt = delta[r * DINNER + d];
        const float uu  = u[r * DINNER + d];
        const float bc  = x_dbl[r * (DTRANK + 2 * DSTATE) + DTRANK + lane];
#pragma unroll
        for (int n = 0; n < DSTATE; ++n) {
            const float Bn = __shfl(bc, n, 32);
            const float dA = __expf(dlt * Arow[n]);
            s[n]  = fmaf(dA, s[n], dlt * Bn * uu);
            aP[n] *= dA;
        }
    }
#pragma unroll
    for (int n = 0; n < DSTATE; ++n) {
        const size_t idx = (((size_t)(b * NCHUNK + c)) * DSTATE + n) * DINNER + d;
        chunkA[idx] = aP[n];
        chunkS[idx] = s[n];
    }
}

__global__ void scan_phase_b(const float* __restrict__ chunkA,
                             const float* __restrict__ chunkS,
                             float* __restrict__ initS) {
    const int g = blockIdx.x * blockDim.x + threadIdx.x;  // 0..8191
    const int d = g & (DINNER - 1);
    const int b = g >> 11;

    float s[DSTATE];
#pragma unroll
    for (int n = 0; n < DSTATE; ++n) s[n] = 0.f;

    for (int c = 0; c < NCHUNK; ++c) {
#pragma unroll
        for (int n = 0; n < DSTATE; ++n) {
            const size_t idx = (((size_t)(b * NCHUNK + c)) * DSTATE + n) * DINNER + d;
            initS[idx] = s[n];
            s[n] = fmaf(chunkA[idx], s[n], chunkS[idx]);
        }
    }
}

__global__ void scan_phase_c(float* __restrict__ u_y,
                             const float* __restrict__ delta,
                             const float* __restrict__ x_dbl,
                             const float* __restrict__ res,
                             const float* __restrict__ A_log,
                             const float* __restrict__ Dp,
                             const float* __restrict__ initS) {
    const int g = blockIdx.x * blockDim.x + threadIdx.x;  // 0..524287
    const int d = g & (DINNER - 1);
    const int rest = g >> 11;
    const int c = rest & (NCHUNK - 1);
    const int b = rest >> 6;
    const int lane = threadIdx.x & 31;

    float Arow[DSTATE], s[DSTATE];
#pragma unroll
    for (int n = 0; n < DSTATE; ++n) {
        Arow[n] = -__expf(A_log[d * DSTATE + n]);
        const size_t idx = (((size_t)(b * NCHUNK + c)) * DSTATE + n) * DINNER + d;
        s[n] = initS[idx];
    }
    const float Dv = Dp[d];

    const size_t rowBase = (size_t)b * LSEQ + (size_t)c * CLEN;
    for (int l0 = 0; l0 < CLEN; ++l0) {
        const size_t r = rowBase + l0;
        const float dlt = delta[r * DINNER + d];
        const float uu  = u_y[r * DINNER + d];
        const float bc  = x_dbl[r * (DTRANK + 2 * DSTATE) + DTRANK + lane];

        float y = 0.f;
#pragma unroll
        for (int n = 0; n < DSTATE; ++n) {
            const float Bn = __shfl(bc, n, 32);
            const float Cn = __shfl(bc, n + DSTATE, 32);
            const float dA = __expf(dlt * Arow[n]);
            s[n] = fmaf(dA, s[n], dlt * Bn * uu);
            y = fmaf(s[n], Cn, y);
        }
        const float rv = res[r * DINNER + d];
        u_y[r * DINNER + d] = fmaf(uu, Dv, y) * silu_f(rv);
    }
}

// ---------------------------------------------------------------------------
extern "C" void kernel_launch(void* const* d_in, const int* in_sizes, int n_in,
                              void* d_out, int out_size, void* d_ws, size_t ws_size,
                              hipStream_t stream) {
    (void)in_sizes; (void)n_in; (void)out_size; (void)ws_size;

    const float* x      = (const float*)d_in[0];  // (4,4096,1024)
    const float* W_in   = (const float*)d_in[1];  // (4096,1024)
    const float* conv_w = (const float*)d_in[2];  // (2048,1,4)
    const float* conv_b = (const float*)d_in[3];  // (2048,)
    const float* W_x    = (const float*)d_in[4];  // (96,2048)
    const float* W_dt   = (const float*)d_in[5];  // (2048,64)
    const float* b_dt   = (const float*)d_in[6];  // (2048,)
    const float* A_log  = (const float*)d_in[7];  // (2048,16)
    const float* D_par  = (const float*)d_in[8];  // (2048,)
    const float* W_out  = (const float*)d_in[9];  // (1024,2048)
    float* out = (float*)d_out;                   // (4,4096,1024)

    // workspace
    float* xh_pre = (float*)d_ws;                       // (B,DINNER,L); later delta
    float* res    = xh_pre + (size_t)MROWS * DINNER;
    float* xh_act = res    + (size_t)MROWS * DINNER;    // becomes y_gated in place
    float* x_dbl  = xh_act + (size_t)MROWS * DINNER;    // (B*L,96)
    float* chunkA = x_dbl  + (size_t)MROWS * (DTRANK + 2 * DSTATE);
    float* chunkS = chunkA + (size_t)BSZ * NCHUNK * DSTATE * DINNER;
    float* initS  = chunkS + (size_t)BSZ * NCHUNK * DSTATE * DINNER;
    float* deltaB = xh_pre;                             // reuse after conv

    // 1) xz = x @ W_in^T, split -> xh_pre (transposed) + res
    wmma_gemm_kernel<1, 2, 2><<<dim3((2 * DINNER) / 64, MROWS / 128), 256, 0, stream>>>(
        x, W_in, xh_pre, res, nullptr,
        MROWS, 2 * DINNER, DMODEL, DMODEL, DMODEL, 0);

    // 2) depthwise causal conv + SiLU -> xh_act (B,L,DINNER)
    conv_silu_kernel<<<((size_t)BSZ * DINNER * LSEQ) / 256, 256, 0, stream>>>(
        xh_pre, conv_w, conv_b, xh_act);

    // 3) x_dbl = xh_act @ W_x^T   (N = 96 -> TN=1)
    wmma_gemm_kernel<0, 2, 1><<<dim3(96 / 32, MROWS / 128), 256, 0, stream>>>(
        xh_act, W_x, x_dbl, nullptr, nullptr,
        MROWS, 96, DINNER, DINNER, DINNER, 96);

    // 4) delta = softplus(dt_r @ W_dt^T + 2*b_dt)  (reference adds b_dt twice)
    wmma_gemm_kernel<2, 2, 2><<<dim3(DINNER / 64, MROWS / 128), 256, 0, stream>>>(
        x_dbl, W_dt, deltaB, nullptr, b_dt,
        MROWS, DINNER, DTRANK, DTRANK + 2 * DSTATE, DTRANK, DINNER);

    // 5) chunked parallel scan + skip + gating, in place on xh_act
    scan_phase_a<<<(BSZ * NCHUNK * DINNER) / 256, 256, 0, stream>>>(
        xh_act, deltaB, x_dbl, A_log, chunkA, chunkS);
    scan_phase_b<<<(BSZ * DINNER) / 256, 256, 0, stream>>>(
        chunkA, chunkS, initS);
    scan_phase_c<<<(BSZ * NCHUNK * DINNER) / 256, 256, 0, stream>>>(
        xh_act, deltaB, x_dbl, res, A_log, D_par, initS);

    // 6) out = y_gated @ W_out^T
    wmma_gemm_kernel<0, 2, 2><<<dim3(DMODEL / 64, MROWS / 128), 256, 0, stream>>>(
        xh_act, W_out, out, nullptr, nullptr,
        MROWS, DMODEL, DINNER, DINNER, DINNER, DMODEL);
}